// Molmo2Attention_21423296873056
// MI455X (gfx1250) — compile-verified
//
#include <hip/hip_runtime.h>
#include <hip/hip_bf16.h>
#include <math.h>

// ---------------------------------------------------------------------------
// Molmo2 attention layer on gfx1250 (MI455X), fp32 WMMA path.
//   S=2048, HID=3584, H=28, KH=4, D=128, G=7
// Round 2: register double-buffering so WMMA chains overlap in-flight loads
// instead of load -> s_wait_loadcnt 0 -> wmma serialization.
// ---------------------------------------------------------------------------

#define S_TOK   2048
#define HID     3584
#define NH      28
#define NKH     4
#define HD      128
#define GRP     7                 // NH / NKH
#define QKV_N   4608              // (H + 2*KH) * D
#define QDIM    3584              // H*D
#define KDIM    512               // KH*D
#define EPS     1e-6f
#define SCALE   0.08838834764831845f   // D^-0.5
// ln(1e6)/64  (inv_freq[d] = exp(-d * this))
#define ROPE_C  0.21586735246819178f

typedef __attribute__((ext_vector_type(2))) float v2f;
typedef __attribute__((ext_vector_type(8))) float v8f;

static __device__ __forceinline__ v8f wmma_f32(v2f a, v2f b, v8f c) {
    // D = A(16x4 f32) * B(4x16 f32) + C(16x16 f32)
    return __builtin_amdgcn_wmma_f32_16x16x4_f32(
        /*neg_a=*/false, a, /*neg_b=*/false, b,
        /*c_mod=*/(short)0, c, /*reuse_a=*/false, /*reuse_b=*/false);
}

// ---------------------------------------------------------------------------
// GEMM: C[M x N] = A[M x K] * B[N x K]^T (+ bias[N]), all row-major fp32.
// Wave computes a 32x64 strip: 2 A fragments x 4 B fragments = 8 WMMA per
// 6 b64 loads. k-chunks of 4 are double-buffered in registers: next chunk's
// loads are issued before the current chunk's WMMAs.
// grid.x = M/32, grid.y = N/512, block = 256 (8 waves).
// A fragment (16x4 f32): a.x = A[m][k + 2*hi], a.y = A[m][k + 2*hi + 1]
// B fragment (4x16 f32): b.x = B[n][k + 2*hi], b.y = B[n][k + 2*hi + 1]
// ---------------------------------------------------------------------------
__global__ __launch_bounds__(256)
void gemm_nt(const float* __restrict__ A, const float* __restrict__ B,
             const float* __restrict__ bias, float* __restrict__ C,
             int K, int ldc) {
    const int lane = threadIdx.x & 31;
    const int wave = threadIdx.x >> 5;
    const int lo = lane & 15;
    const int hi = lane >> 4;

    const int m0 = blockIdx.x * 32;
    const int n0 = (blockIdx.y * 8 + wave) * 64;

    v8f acc[2][4];
    #pragma unroll
    for (int j = 0; j < 4; ++j) {
        float bv = bias ? bias[n0 + j * 16 + lo] : 0.0f;
        v8f t = {bv, bv, bv, bv, bv, bv, bv, bv};
        acc[0][j] = t;
        acc[1][j] = t;
    }

    const unsigned a0off = (unsigned)(m0 + lo) * K + 2 * hi;
    const unsigned a1off = a0off + 16u * K;
    const unsigned boff  = (unsigned)(n0 + lo) * K + 2 * hi;

    // prologue: load k-chunk 0
    v2f a0c = *(const v2f*)(A + a0off);
    v2f a1c = *(const v2f*)(A + a1off);
    v2f bc[4];
    #pragma unroll
    for (int j = 0; j < 4; ++j)
        bc[j] = *(const v2f*)(B + boff + (unsigned)j * 16u * K);

    #pragma unroll 4
    for (int k = 4; k < K; k += 4) {
        // issue next chunk's loads first ...
        v2f a0n = *(const v2f*)(A + a0off + k);
        v2f a1n = *(const v2f*)(A + a1off + k);
        v2f bn[4];
        #pragma unroll
        for (int j = 0; j < 4; ++j)
            bn[j] = *(const v2f*)(B + boff + (unsigned)j * 16u * K + k);
        // ... then consume the current chunk
        #pragma unroll
        for (int j = 0; j < 4; ++j) acc[0][j] = wmma_f32(a0c, bc[j], acc[0][j]);
        #pragma unroll
        for (int j = 0; j < 4; ++j) acc[1][j] = wmma_f32(a1c, bc[j], acc[1][j]);
        a0c = a0n; a1c = a1n;
        #pragma unroll
        for (int j = 0; j < 4; ++j) bc[j] = bn[j];
    }
    // epilogue chunk
    #pragma unroll
    for (int j = 0; j < 4; ++j) acc[0][j] = wmma_f32(a0c, bc[j], acc[0][j]);
    #pragma unroll
    for (int j = 0; j < 4; ++j) acc[1][j] = wmma_f32(a1c, bc[j], acc[1][j]);

    // C layout: vgpr r -> row (r + 8*hi), lane lo -> col
    #pragma unroll
    for (int mi = 0; mi < 2; ++mi)
        #pragma unroll
        for (int j = 0; j < 4; ++j)
            #pragma unroll
            for (int r = 0; r < 8; ++r)
                C[(m0 + mi * 16 + r + 8 * hi) * ldc + n0 + j * 16 + lo] = acc[mi][j][r];
}

// ---------------------------------------------------------------------------
// RMSNorm (q over 3584, k over 512) + RoPE. One block (256 thr) per token.
// qkv row layout: [ q(3584) | k(512) | v(512) ]
// ---------------------------------------------------------------------------
__global__ __launch_bounds__(256)
void rmsnorm_rope(const float* __restrict__ qkv, const int* __restrict__ pos,
                  const float* __restrict__ qw, const float* __restrict__ kw,
                  float* __restrict__ qout, float* __restrict__ kout) {
    __shared__ float red[256];
    const int s   = blockIdx.x;
    const int tid = threadIdx.x;
    const float* row = qkv + s * QKV_N;

    // --- q rms ---
    float sq = 0.0f;
    for (int i = tid; i < QDIM; i += 256) { float x = row[i]; sq += x * x; }
    red[tid] = sq; __syncthreads();
    for (int off = 128; off > 0; off >>= 1) {
        if (tid < off) red[tid] += red[tid + off];
        __syncthreads();
    }
    const float qrms = rsqrtf(red[0] / (float)QDIM + EPS);
    __syncthreads();

    // --- k rms ---
    float sk = 0.0f;
    for (int i = tid; i < KDIM; i += 256) { float x = row[QDIM + i]; sk += x * x; }
    red[tid] = sk; __syncthreads();
    for (int off = 128; off > 0; off >>= 1) {
        if (tid < off) red[tid] += red[tid + off];
        __syncthreads();
    }
    const float krms = rsqrtf(red[0] / (float)KDIM + EPS);

    const float p = (float)pos[s];

    // --- q: H*64 rotation pairs ---
    for (int idx = tid; idx < NH * 64; idx += 256) {
        const int h = idx >> 6;
        const int d = idx & 63;
        const float fr = p * expf(-(float)d * ROPE_C);
        const float c = cosf(fr), sn = sinf(fr);
        const int base = h * HD + d;
        const float x1 = row[base]      * qrms * qw[base];
        const float x2 = row[base + 64] * qrms * qw[base + 64];
        qout[s * QDIM + base]      = x1 * c - x2 * sn;
        qout[s * QDIM + base + 64] = x2 * c + x1 * sn;
    }
    // --- k: KH*64 = 256 pairs, one per thread ---
    {
        const int h = tid >> 6;
        const int d = tid & 63;
        const float fr = p * expf(-(float)d * ROPE_C);
        const float c = cosf(fr), sn = sinf(fr);
        const int base = h * HD + d;
        const float x1 = row[QDIM + base]      * krms * kw[base];
        const float x2 = row[QDIM + base + 64] * krms * kw[base + 64];
        kout[s * KDIM + base]      = x1 * c - x2 * sn;
        kout[s * KDIM + base + 64] = x2 * c + x1 * sn;
    }
}

// ---------------------------------------------------------------------------
// Flash attention, one wave per (head h, 16-row q-tile).
// grid = (28, 32), block = 128 (4 waves -> 4 consecutive q-tiles).
// QK^T: 32 wmma over D=128, K fragments ping-pong buffered in chunks of 8.
// PV: probs bounced through LDS (C->A relayout), V fragments ping-pong
// buffered per 16-wide d-tile (prefetch j+1 while multiplying j).
// Online softmax via __shfl_xor over the 16-lane halves (C layout: N in
// lanes, M striped over vgpr index / wave half).
// ---------------------------------------------------------------------------
__global__ __launch_bounds__(128)
void attn_kernel(const float* __restrict__ qr, const float* __restrict__ kr,
                 const float* __restrict__ qkv, float* __restrict__ attn) {
    __shared__ float plds[4][256];

    const int lane = threadIdx.x & 31;
    const int wave = threadIdx.x >> 5;
    const int lo = lane & 15;
    const int hi = lane >> 4;

    const int h  = blockIdx.x;        // q head
    const int kv = h / GRP;           // kv head
    const int s0 = (blockIdx.y * 4 + wave) * 16;

    const float* Qp = qr + h * HD;                  // row stride QDIM
    const float* Kp = kr + kv * HD;                 // row stride KDIM
    const float* Vp = qkv + QDIM + KDIM + kv * HD;  // row stride QKV_N

    // Preload Q tile fragments: qa[kk].x = Q[s0+lo][4*kk + 2*hi], .y = +1
    v2f qa[32];
    {
        const float* qrow = Qp + (s0 + lo) * QDIM + 2 * hi;
        #pragma unroll
        for (int kk = 0; kk < 32; ++kk)
            qa[kk] = *(const v2f*)(qrow + kk * 4);
    }

    v8f acc[8];
    #pragma unroll
    for (int j = 0; j < 8; ++j) {
        v8f z = {0, 0, 0, 0, 0, 0, 0, 0};
        acc[j] = z;
    }
    float m[8], l[8];
    #pragma unroll
    for (int r = 0; r < 8; ++r) { m[r] = -3.0e38f; l[r] = 0.0f; }

    float* P = plds[wave];

    for (int t0 = 0; t0 <= s0; t0 += 16) {
        // ---- scores = Q K^T, double-buffered in chunks of 8 fragments ----
        v8f sc = {0, 0, 0, 0, 0, 0, 0, 0};
        const float* kb = Kp + (t0 + lo) * KDIM + 2 * hi;

        v2f kbuf[2][8];
        #pragma unroll
        for (int i = 0; i < 8; ++i)
            kbuf[0][i] = *(const v2f*)(kb + i * 4);
        #pragma unroll
        for (int c = 0; c < 4; ++c) {
            if (c < 3) {
                #pragma unroll
                for (int i = 0; i < 8; ++i)
                    kbuf[(c + 1) & 1][i] = *(const v2f*)(kb + (c + 1) * 32 + i * 4);
            }
            #pragma unroll
            for (int i = 0; i < 8; ++i)
                sc = wmma_f32(qa[c * 8 + i], kbuf[c & 1][i], sc);
        }

        #pragma unroll
        for (int r = 0; r < 8; ++r) sc[r] *= SCALE;

        if (t0 == s0) {   // causal mask on the diagonal tile: keep row >= col
            #pragma unroll
            for (int r = 0; r < 8; ++r)
                if ((r + 8 * hi) < lo) sc[r] = -3.0e38f;
        }

        // ---- row max (reduce over 16-lane group = one score row) ----
        float tm[8];
        #pragma unroll
        for (int r = 0; r < 8; ++r) tm[r] = sc[r];
        #pragma unroll
        for (int mk = 1; mk < 16; mk <<= 1)
            #pragma unroll
            for (int r = 0; r < 8; ++r)
                tm[r] = fmaxf(tm[r], __shfl_xor(tm[r], mk, 32));

        // ---- online softmax update ----
        float pr[8];
        #pragma unroll
        for (int r = 0; r < 8; ++r) {
            const float nm   = fmaxf(m[r], tm[r]);
            const float corr = __expf(m[r] - nm);
            m[r]  = nm;
            pr[r] = __expf(sc[r] - nm);
            l[r] *= corr;
            #pragma unroll
            for (int j = 0; j < 8; ++j) acc[j][r] *= corr;
        }
        float rs[8];
        #pragma unroll
        for (int r = 0; r < 8; ++r) rs[r] = pr[r];
        #pragma unroll
        for (int mk = 1; mk < 16; mk <<= 1)
            #pragma unroll
            for (int r = 0; r < 8; ++r)
                rs[r] += __shfl_xor(rs[r], mk, 32);
        #pragma unroll
        for (int r = 0; r < 8; ++r) l[r] += rs[r];

        // ---- probs: C layout -> LDS -> A layout (same-wave, wait dscnt) ----
        #pragma unroll
        for (int r = 0; r < 8; ++r)
            P[(r + 8 * hi) * 16 + lo] = pr[r];
        asm volatile("s_wait_dscnt 0x0" ::: "memory");

        v2f pa[4];
        #pragma unroll
        for (int kk = 0; kk < 4; ++kk)
            pa[kk] = *(const v2f*)(P + lo * 16 + kk * 4 + 2 * hi);

        // ---- acc += P V, ping-pong V fragments across d-tiles ----
        const float* vb = Vp + t0 * QKV_N + lo;
        v2f vbuf[2][4];
        #pragma unroll
        for (int kk = 0; kk < 4; ++kk) {
            vbuf[0][kk].x = vb[(kk * 4 + 2 * hi)     * QKV_N];
            vbuf[0][kk].y = vb[(kk * 4 + 2 * hi + 1) * QKV_N];
        }
        #pragma unroll
        for (int j = 0; j < 8; ++j) {
            if (j < 7) {
                #pragma unroll
                for (int kk = 0; kk < 4; ++kk) {
                    vbuf[(j + 1) & 1][kk].x = vb[(j + 1) * 16 + (kk * 4 + 2 * hi)     * QKV_N];
                    vbuf[(j + 1) & 1][kk].y = vb[(j + 1) * 16 + (kk * 4 + 2 * hi + 1) * QKV_N];
                }
            }
            #pragma unroll
            for (int kk = 0; kk < 4; ++kk)
                acc[j] = wmma_f32(pa[kk], vbuf[j & 1][kk], acc[j]);
        }
    }

    // ---- finalize: attn[s][h*128 + d] = acc / l ----
    #pragma unroll
    for (int r = 0; r < 8; ++r) {
        const float inv_l = 1.0f / l[r];
        const int   orow  = (s0 + r + 8 * hi) * QDIM + h * HD + lo;
        #pragma unroll
        for (int j = 0; j < 8; ++j)
            attn[orow + j * 16] = acc[j][r] * inv_l;
    }
}

// ---------------------------------------------------------------------------
// Launch: qkv GEMM -> rmsnorm+rope -> flash attention -> output GEMM
// ---------------------------------------------------------------------------
extern "C" void kernel_launch(void* const* d_in, const int* in_sizes, int n_in,
                              void* d_out, int out_size, void* d_ws, size_t ws_size,
                              hipStream_t stream) {
    const int*   positions = (const int*)  d_in[0];
    const float* hidden    = (const float*)d_in[1];
    const float* qkv_w     = (const float*)d_in[2];
    const float* qkv_b     = (const float*)d_in[3];
    const float* q_norm_w  = (const float*)d_in[4];
    const float* k_norm_w  = (const float*)d_in[5];
    const float* o_w       = (const float*)d_in[6];
    float*       out       = (float*)d_out;

    // Workspace layout (floats): qkv | q_rope | k_rope | attn
    float* ws      = (float*)d_ws;
    float* qkv     = ws;                                   // S * 4608
    float* q_rope  = qkv    + (size_t)S_TOK * QKV_N;       // S * 3584
    float* k_rope  = q_rope + (size_t)S_TOK * QDIM;        // S * 512
    float* attn_ws = k_rope + (size_t)S_TOK * KDIM;        // S * 3584

    // 1) qkv = hidden @ qkv_w^T + qkv_b   (M=2048, N=4608, K=3584)
    gemm_nt<<<dim3(S_TOK / 32, QKV_N / 512), 256, 0, stream>>>(
        hidden, qkv_w, qkv_b, qkv, HID, QKV_N);

    // 2) rmsnorm + rope
    rmsnorm_rope<<<S_TOK, 256, 0, stream>>>(
        qkv, positions, q_norm_w, k_norm_w, q_rope, k_rope);

    // 3) flash attention
    attn_kernel<<<dim3(NH, S_TOK / 16 / 4), 128, 0, stream>>>(
        q_rope, k_rope, qkv, attn_ws);

    // 4) out = attn @ o_w^T   (M=2048, N=3584, K=3584)
    gemm_nt<<<dim3(S_TOK / 32, HID / 512), 256, 0, stream>>>(
        attn_ws, o_w, nullptr, out, QDIM, HID);
}